// GraphODEFunc_14594298872163
// MI455X (gfx1250) — compile-verified
//
#include <hip/hip_runtime.h>

#define N_NODES 100000
#define N_EDGES 1600000
#define DIM 128

typedef __attribute__((ext_vector_type(2))) float v2f;
typedef __attribute__((ext_vector_type(8))) float v8f;

// ---------------------------------------------------------------------------
// Degree: deg[dst] += 1 per edge (hardware fp32 atomic)
// ---------------------------------------------------------------------------
__global__ __launch_bounds__(256) void deg_kernel(const int* __restrict__ dst,
                                                  float* __restrict__ deg) {
    int e = blockIdx.x * blockDim.x + threadIdx.x;
    if (e < N_EDGES) {
        unsafeAtomicAdd(&deg[dst[e]], 1.0f);
    }
}

// deg -> deg_inv in place: deg>0 ? 1/deg : 0
__global__ __launch_bounds__(256) void deginv_kernel(float* __restrict__ deg) {
    int i = blockIdx.x * blockDim.x + threadIdx.x;
    if (i < N_NODES) {
        float d = deg[i];
        deg[i] = (d > 0.0f) ? (1.0f / d) : 0.0f;
    }
}

// ---------------------------------------------------------------------------
// Scatter: one wave32 per edge. Each lane gathers float4 of x[src] and does
// 4 fp32 global atomic adds into agg[dst]. 8 edges per 256-thread block.
// ---------------------------------------------------------------------------
__global__ __launch_bounds__(256) void scatter_kernel(const float* __restrict__ x,
                                                      const int* __restrict__ src,
                                                      const int* __restrict__ dst,
                                                      float* __restrict__ agg) {
    int e    = (int)((blockIdx.x * (unsigned)blockDim.x + threadIdx.x) >> 5);
    int lane = threadIdx.x & 31;
    if (e >= N_EDGES) return;
    int s = src[e];
    int d = dst[e];
    const float4 v = *(const float4*)(x + (size_t)s * DIM + lane * 4);
    float* ap = agg + (size_t)d * DIM + lane * 4;
    unsafeAtomicAdd(ap + 0, v.x);
    unsafeAtomicAdd(ap + 1, v.y);
    unsafeAtomicAdd(ap + 2, v.z);
    unsafeAtomicAdd(ap + 3, v.w);
}

// ---------------------------------------------------------------------------
// Fused SAGE GEMM epilogue:
//   out = relu?( (agg*deg_inv) @ Wl + xin @ Wr + b )
// Block: 256 threads = 8 waves; wave w owns output cols [w*16, w*16+16).
// Each wave holds TWO 16x16 fp32 accumulators (rows rowbase..+15, +16..+31),
// reusing the Wl/Wr B-fragments across both row tiles.
// Uses V_WMMA_F32_16X16X4_F32 (fp32 WMMA, K=4 per step).
// ---------------------------------------------------------------------------
__global__ __launch_bounds__(256) void sage_gemm_kernel(
        const float* __restrict__ agg, const float* __restrict__ deg_inv,
        const float* __restrict__ xin,
        const float* __restrict__ Wl, const float* __restrict__ Wr,
        const float* __restrict__ bias,
        float* __restrict__ out, int relu) {
    const int rowbase = blockIdx.x * 32;          // 32 rows per block
    const int wave    = threadIdx.x >> 5;         // 0..7 -> N tile
    const int lane    = threadIdx.x & 31;
    const int m       = lane & 15;                // A row within tile / B col
    const int kb      = (lane >> 4) << 1;         // 0 or 2 (K sub-offset)
    const int ncol    = wave * 16 + m;            // output / weight column

    const int row0 = rowbase + m;                 // row tile 0
    const int row1 = rowbase + 16 + m;            // row tile 1
    const float dinv0 = deg_inv[row0];
    const float dinv1 = deg_inv[row1];

    const float* aggr0 = agg + (size_t)row0 * DIM;
    const float* aggr1 = agg + (size_t)row1 * DIM;
    const float* xr0   = xin + (size_t)row0 * DIM;
    const float* xr1   = xin + (size_t)row1 * DIM;

    v8f c0 = {};
    v8f c1 = {};

    #pragma unroll 4
    for (int k = 0; k < DIM; k += 4) {
        // B fragments (shared across both row tiles):
        // lanes 0-15 hold K=k+kb (VGPR0) and K=k+kb+1 (VGPR1), N=lane&15.
        v2f bl, br;
        bl.x = Wl[(size_t)(k + kb)     * DIM + ncol];
        bl.y = Wl[(size_t)(k + kb + 1) * DIM + ncol];
        br.x = Wr[(size_t)(k + kb)     * DIM + ncol];
        br.y = Wr[(size_t)(k + kb + 1) * DIM + ncol];

        // A fragments: contiguous float2 per lane at K = k+kb.
        float2 t0 = *(const float2*)(aggr0 + k + kb);
        float2 u0 = *(const float2*)(xr0   + k + kb);
        float2 t1 = *(const float2*)(aggr1 + k + kb);
        float2 u1 = *(const float2*)(xr1   + k + kb);
        v2f am0; am0.x = t0.x * dinv0; am0.y = t0.y * dinv0;
        v2f ax0; ax0.x = u0.x;         ax0.y = u0.y;
        v2f am1; am1.x = t1.x * dinv1; am1.y = t1.y * dinv1;
        v2f ax1; ax1.x = u1.x;         ax1.y = u1.y;

        c0 = __builtin_amdgcn_wmma_f32_16x16x4_f32(false, am0, false, bl,
                                                   (short)0, c0, false, false);
        c0 = __builtin_amdgcn_wmma_f32_16x16x4_f32(false, ax0, false, br,
                                                   (short)0, c0, false, false);
        c1 = __builtin_amdgcn_wmma_f32_16x16x4_f32(false, am1, false, bl,
                                                   (short)0, c1, false, false);
        c1 = __builtin_amdgcn_wmma_f32_16x16x4_f32(false, ax1, false, br,
                                                   (short)0, c1, false, false);
    }

    // Epilogue: C/D layout — VGPR r: lanes 0-15 -> M=r, lanes 16-31 -> M=r+8.
    const float bv   = bias[ncol];
    const int   mofs = (lane >> 4) * 8;
    #pragma unroll
    for (int r = 0; r < 8; ++r) {
        float v0 = c0[r] + bv;
        float v1 = c1[r] + bv;
        if (relu) { v0 = fmaxf(v0, 0.0f); v1 = fmaxf(v1, 0.0f); }
        out[(size_t)(rowbase + mofs + r) * DIM + ncol]      = v0;
        out[(size_t)(rowbase + 16 + mofs + r) * DIM + ncol] = v1;
    }
}

// ---------------------------------------------------------------------------
extern "C" void kernel_launch(void* const* d_in, const int* in_sizes, int n_in,
                              void* d_out, int out_size, void* d_ws, size_t ws_size,
                              hipStream_t stream) {
    // setup_inputs order: t, x, edge_index, W1_l, W1_r, b1, W2_l, W2_r, b2,
    //                     W3_l, W3_r, b3
    const float* x   = (const float*)d_in[1];
    const int*   ei  = (const int*)d_in[2];
    const int*   src = ei;                 // edge_index[0]
    const int*   dst = ei + N_EDGES;       // edge_index[1]
    const float* W1l = (const float*)d_in[3];
    const float* W1r = (const float*)d_in[4];
    const float* b1  = (const float*)d_in[5];
    const float* W2l = (const float*)d_in[6];
    const float* W2r = (const float*)d_in[7];
    const float* b2  = (const float*)d_in[8];
    const float* W3l = (const float*)d_in[9];
    const float* W3r = (const float*)d_in[10];
    const float* b3  = (const float*)d_in[11];
    float* outp = (float*)d_out;

    // Workspace: deg_inv[N] | agg[N*128] | h1[N*128] | h2[N*128]  (~154 MB)
    float* deg = (float*)d_ws;
    float* agg = deg + N_NODES;
    float* h1  = agg + (size_t)N_NODES * DIM;
    float* h2  = h1  + (size_t)N_NODES * DIM;

    const size_t aggBytes = (size_t)N_NODES * DIM * sizeof(float);
    const int degBlocks   = (N_EDGES + 255) / 256;
    const int nodeBlocks  = (N_NODES + 255) / 256;
    const int scatBlocks  = (N_EDGES + 7) / 8;     // 8 edges (waves) per block
    const int gemmBlocks  = N_NODES / 32;          // 3125, exact

    // Degrees -> deg_inv (in place)
    hipMemsetAsync(deg, 0, N_NODES * sizeof(float), stream);
    deg_kernel<<<degBlocks, 256, 0, stream>>>(dst, deg);
    deginv_kernel<<<nodeBlocks, 256, 0, stream>>>(deg);

    // Layer 1: x -> h1 (relu)
    hipMemsetAsync(agg, 0, aggBytes, stream);
    scatter_kernel<<<scatBlocks, 256, 0, stream>>>(x, src, dst, agg);
    sage_gemm_kernel<<<gemmBlocks, 256, 0, stream>>>(agg, deg, x, W1l, W1r, b1, h1, 1);

    // Layer 2: h1 -> h2 (relu)
    hipMemsetAsync(agg, 0, aggBytes, stream);
    scatter_kernel<<<scatBlocks, 256, 0, stream>>>(h1, src, dst, agg);
    sage_gemm_kernel<<<gemmBlocks, 256, 0, stream>>>(agg, deg, h1, W2l, W2r, b2, h2, 1);

    // Layer 3: h2 -> d_out (no relu)
    hipMemsetAsync(agg, 0, aggBytes, stream);
    scatter_kernel<<<scatBlocks, 256, 0, stream>>>(h2, src, dst, agg);
    sage_gemm_kernel<<<gemmBlocks, 256, 0, stream>>>(agg, deg, h2, W3l, W3r, b3, outp, 0);
}